// TransformerBlockQuantum_65481071407857
// MI455X (gfx1250) — compile-verified
//
#include <hip/hip_runtime.h>

// ---------------------------------------------------------------------------
// Fused quantum transformer block for MI455X (gfx1250, wave32, WMMA).
//
// Key simplifications proven from the reference:
//  * S == 1  =>  softmax over a single key is identically 1, so the attention
//    output equals V = LN1(x) @ wv.T exactly. wq/wk are dead.
//  * The CNOT-ring "quantum layer" is linear over GF(2); for a product input
//    distribution each Z-expectation factorizes:
//       out[w] = prod_{v=0..w} cos(x_v + th_v)  (w >= 1)
//       out[0] = prod_{v=1..7} cos(x_v + th_v)
//
// GEMMs (128x128 wv, 128x128 wc, 512->128 w2) run on V_WMMA_F32_16X16X4_F32
// (fp32 A/B, fp32 accum -> same precision class as the fp32 reference).
// New this round: x-tile staged via GLOBAL_LOAD_ASYNC_TO_LDS_B128 (ASYNCcnt
// path, no VGPR round-trip) and global_prefetch_b8 on the w2 stream.
// ---------------------------------------------------------------------------

typedef __attribute__((ext_vector_type(2))) float v2f;
typedef __attribute__((ext_vector_type(8))) float v8f;

#define EDIM 128
#define FDIM 512
#define ROWS 32          // rows per workgroup
#define LDSP 132         // padded LDS row stride (floats)
#define NBLK (8192 / ROWS)

__device__ __forceinline__ v8f wmma4_f32(v2f a, v2f b, v8f c) {
  // (neg_a, A, neg_b, B, c_mod, C, reuse_a, reuse_b)
  return __builtin_amdgcn_wmma_f32_16x16x4_f32(false, a, false, b, (short)0, c,
                                               false, false);
}

// closed-form ring-CNOT qlayer on 8 wires
__device__ __forceinline__ void qlayer8(const float* __restrict__ in,
                                        const float* __restrict__ th,
                                        float* __restrict__ out) {
  float c[8];
#pragma unroll
  for (int w = 0; w < 8; ++w) c[w] = cosf(in[w] + th[w]);
  float p = c[0];
#pragma unroll
  for (int w = 1; w < 8; ++w) { p *= c[w]; out[w] = p; }
  float s = c[1];
#pragma unroll
  for (int w = 2; w < 8; ++w) s *= c[w];
  out[0] = s;
}

__global__ __launch_bounds__(256)
void qtblock_kernel(const float* __restrict__ x,
                    const float* __restrict__ ln1_g, const float* __restrict__ ln1_b,
                    const float* __restrict__ ln2_g, const float* __restrict__ ln2_b,
                    const float* __restrict__ wv, const float* __restrict__ wc,
                    const float* __restrict__ bc,
                    const float* __restrict__ th_a, const float* __restrict__ th_f,
                    const float* __restrict__ w1, const float* __restrict__ b1,
                    const float* __restrict__ w2, const float* __restrict__ b2,
                    float* __restrict__ out) {
  __shared__ float sx[ROWS][LDSP];   // x tile, becomes x2 (residual stream)
  __shared__ float sh[ROWS][LDSP];   // LN1 out (h); later ReLU(a1) chunks
  __shared__ float st[ROWS][LDSP];   // V; then qh (in-place)
  __shared__ float sqf[ROWS][8];     // qlayer(h2[:, :8])

  const int tid  = threadIdx.x;
  const int wave = tid >> 5;
  const int lane = tid & 31;
  const int lm   = lane & 15;        // n (or m) within a 16x16 tile
  const int lhi  = lane >> 4;        // 0/1 half-wave selector
  const int lk   = lhi << 1;         // K sub-offset for A/B fragments
  const int mt   = wave & 1;         // M tile (2 tiles of 16 rows)
  const int nt0  = (wave >> 1) << 1; // N tile base (2 tiles per wave)
  const long rowbase = (long)blockIdx.x * ROWS;

  // ---- stage 0: async global->LDS copy of the x tile (ASYNCcnt path) ----
  // 16 B per lane per issue; 4 issues per thread cover 32x128 f32.
  // Low 32 bits of a generic pointer into __shared__ are the LDS byte offset.
#pragma unroll
  for (int it = 0; it < 4; ++it) {
    const int i = tid + it * 256;               // i in [0, 1024)
    const int r = i >> 5, c4 = i & 31;
    const float* gp = x + (rowbase + r) * EDIM + c4 * 4;
    const unsigned ldsoff = (unsigned)(size_t)(&sx[r][c4 * 4]);
    asm volatile("global_load_async_to_lds_b128 %0, %1, off"
                 :: "v"(ldsoff), "v"((unsigned long long)(size_t)gp)
                 : "memory");
  }
  asm volatile("s_wait_asynccnt 0" ::: "memory");
  __syncthreads();

  // ---- stage 1: LN1 (one thread per row) --------------------------------
  if (tid < ROWS) {
    float m = 0.f;
    for (int c = 0; c < EDIM; ++c) m += sx[tid][c];
    m *= (1.f / EDIM);
    float v = 0.f;
    for (int c = 0; c < EDIM; ++c) { float d = sx[tid][c] - m; v += d * d; }
    const float inv = rsqrtf(v * (1.f / EDIM) + 1e-5f);
    for (int c = 0; c < EDIM; ++c)
      sh[tid][c] = (sx[tid][c] - m) * inv * ln1_g[c] + ln1_b[c];
  }
  __syncthreads();

  // ---- stage 2: V = h @ wv^T  (WMMA f32 16x16x4) ------------------------
  {
    v8f acc[2] = {};
    for (int ks = 0; ks < EDIM / 4; ++ks) {
      const int k0 = ks * 4;
      v2f a = *reinterpret_cast<const v2f*>(&sh[mt * 16 + lm][k0 + lk]);
#pragma unroll
      for (int j = 0; j < 2; ++j) {
        const float* wp = wv + (size_t)((nt0 + j) * 16 + lm) * EDIM + k0 + lk;
        v2f b = *reinterpret_cast<const v2f*>(wp);
        acc[j] = wmma4_f32(a, b, acc[j]);
      }
    }
#pragma unroll
    for (int j = 0; j < 2; ++j)
#pragma unroll
      for (int r = 0; r < 8; ++r)
        st[mt * 16 + r + lhi * 8][(nt0 + j) * 16 + lm] = acc[j][r];
  }
  __syncthreads();

  // ---- stage 3: per-head qlayer on V (in place, closed form) ------------
  {
    float ta[8];
#pragma unroll
    for (int i = 0; i < 8; ++i) ta[i] = th_a[i];
    for (int t = tid; t < ROWS * 16; t += 256) {
      const int r = t >> 4, h = t & 15;
      float in[8], o[8];
#pragma unroll
      for (int i = 0; i < 8; ++i) in[i] = st[r][h * 8 + i];
      qlayer8(in, ta, o);
#pragma unroll
      for (int i = 0; i < 8; ++i) st[r][h * 8 + i] = o[i];
    }
  }
  __syncthreads();

  // ---- stage 4: x2 = x + qh @ wc^T + bc  (WMMA, add into sx) ------------
  {
    v8f acc[2] = {};
    for (int ks = 0; ks < EDIM / 4; ++ks) {
      const int k0 = ks * 4;
      v2f a = *reinterpret_cast<const v2f*>(&st[mt * 16 + lm][k0 + lk]);
#pragma unroll
      for (int j = 0; j < 2; ++j) {
        const float* wp = wc + (size_t)((nt0 + j) * 16 + lm) * EDIM + k0 + lk;
        v2f b = *reinterpret_cast<const v2f*>(wp);
        acc[j] = wmma4_f32(a, b, acc[j]);
      }
    }
#pragma unroll
    for (int j = 0; j < 2; ++j)
#pragma unroll
      for (int r = 0; r < 8; ++r) {
        const int m = mt * 16 + r + lhi * 8;
        const int n = (nt0 + j) * 16 + lm;
        sx[m][n] = sx[m][n] + acc[j][r] + bc[n];
      }
  }
  __syncthreads();

  // ---- stage 5: LN2 on x2, then qf = qlayer(h2[:, :8]) ------------------
  if (tid < ROWS) {
    float m = 0.f;
    for (int c = 0; c < EDIM; ++c) m += sx[tid][c];
    m *= (1.f / EDIM);
    float v = 0.f;
    for (int c = 0; c < EDIM; ++c) { float d = sx[tid][c] - m; v += d * d; }
    const float inv = rsqrtf(v * (1.f / EDIM) + 1e-5f);
    float h2row[8], tf[8], o[8];
    for (int c = 0; c < EDIM; ++c)
      sh[tid][c] = (sx[tid][c] - m) * inv * ln2_g[c] + ln2_b[c];
#pragma unroll
    for (int i = 0; i < 8; ++i) { h2row[i] = sh[tid][i]; tf[i] = th_f[i]; }
    qlayer8(h2row, tf, o);
#pragma unroll
    for (int i = 0; i < 8; ++i) sqf[tid][i] = o[i];
  }
  __syncthreads();

  // ---- stage 6: FFN = relu(qf @ w1^T + b1) @ w2^T, chunked over F -------
  v8f accF[2] = {};
  for (int fc = 0; fc < 4; ++fc) {
    __syncthreads();  // previous chunk's WMMA reads of sh are done
    // prefetch next chunk's w2 fragments into cache while VALU runs
    if (fc < 3) {
#pragma unroll
      for (int j = 0; j < 2; ++j) {
        const float* wp =
            w2 + (size_t)((nt0 + j) * 16 + lm) * FDIM + (fc + 1) * 128 + lk;
        __builtin_prefetch(wp, 0, 1);
      }
    }
    // a1 chunk (K=8 matvec: VALU is the right tool here)
    for (int i = tid; i < ROWS * 128; i += 256) {
      const int r = i >> 7, fl = i & 127, f = fc * 128 + fl;
      float s = b1[f];
#pragma unroll
      for (int w = 0; w < 8; ++w) s += sqf[r][w] * w1[f * 8 + w];
      sh[r][fl] = fmaxf(s, 0.f);
    }
    __syncthreads();
    // accumulate chunk GEMM: accF += a1_chunk @ w2_chunk^T (WMMA)
    for (int ks = 0; ks < 32; ++ks) {
      const int k0 = ks * 4;
      v2f a = *reinterpret_cast<const v2f*>(&sh[mt * 16 + lm][k0 + lk]);
#pragma unroll
      for (int j = 0; j < 2; ++j) {
        const float* wp =
            w2 + (size_t)((nt0 + j) * 16 + lm) * FDIM + fc * 128 + k0 + lk;
        v2f b = *reinterpret_cast<const v2f*>(wp);
        accF[j] = wmma4_f32(a, b, accF[j]);
      }
    }
  }

  // ---- stage 7: out = x2 + ffn + b2 -------------------------------------
#pragma unroll
  for (int j = 0; j < 2; ++j)
#pragma unroll
    for (int r = 0; r < 8; ++r) {
      const int m = mt * 16 + r + lhi * 8;
      const int n = (nt0 + j) * 16 + lm;
      out[(rowbase + m) * EDIM + n] = sx[m][n] + accF[j][r] + b2[n];
    }
}

extern "C" void kernel_launch(void* const* d_in, const int* in_sizes, int n_in,
                              void* d_out, int out_size, void* d_ws,
                              size_t ws_size, hipStream_t stream) {
  (void)in_sizes; (void)n_in; (void)d_ws; (void)ws_size; (void)out_size;
  const float* x     = (const float*)d_in[0];
  const float* ln1_g = (const float*)d_in[1];
  const float* ln1_b = (const float*)d_in[2];
  const float* ln2_g = (const float*)d_in[3];
  const float* ln2_b = (const float*)d_in[4];
  // d_in[5] = wq, d_in[6] = wk: provably dead (S==1 -> softmax == 1 -> out == V)
  const float* wv    = (const float*)d_in[7];
  const float* wc    = (const float*)d_in[8];
  const float* bc    = (const float*)d_in[9];
  const float* th_a  = (const float*)d_in[10];
  const float* th_f  = (const float*)d_in[11];
  const float* w1    = (const float*)d_in[12];
  const float* b1    = (const float*)d_in[13];
  const float* w2    = (const float*)d_in[14];
  const float* b2    = (const float*)d_in[15];
  float* outp        = (float*)d_out;

  qtblock_kernel<<<NBLK, 256, 0, stream>>>(x, ln1_g, ln1_b, ln2_g, ln2_b, wv,
                                           wc, bc, th_a, th_f, w1, b1, w2, b2,
                                           outp);
}